// BatchSecondOrderLPCSynth_75608604278813
// MI455X (gfx1250) — compile-verified
//
#include <hip/hip_runtime.h>
#include <math.h>
#include <stdint.h>

// ---------------------------------------------------------------------------
// BatchSecondOrderLPCSynth for MI455X (gfx1250, wave32)
//   - one thread per (batch, frame); 11-stage all-pole cascade, *systolic*
//     across stages (stage s processes time i-s) => ILP ~11, latency-immune
//   - block excitation span staged into skewed LDS via async global->LDS
//     copies (global_load_async_to_lds_b32 + s_wait_asynccnt); skew kills
//     the 32-way bank conflict of the 128-float lane stride
//   - overlap-add accumulated in skewed LDS (ds_add_f32), flushed once with
//     coalesced global float atomics; then 1/norm elementwise pass
// ---------------------------------------------------------------------------

#define HOPSZ 128
#define WINSZ 512
#define PADSZ 192                       // (WIN - HOP)/2
#define NSTG  11

#define FPB   32                        // frames per block == threads (1 wave)
#define LDSIN (FPB * HOPSZ + WINSZ - HOPSZ)     // 4480 logical floats
#define LDSSK (LDSIN + LDSIN / HOPSZ)           // 4515 with skew pads

__device__ __forceinline__ int skew(int i) { return i + (i >> 7); }

__device__ __forceinline__ uint32_t lds_byte_off(const void* p) {
    // low 32 bits of a flat LDS pointer == wave-relative LDS byte offset
    return (uint32_t)(uintptr_t)p;
}

__global__ void __launch_bounds__(FPB)
lpc_synth_kernel(const float* __restrict__ ex,
                 const float* __restrict__ gain,
                 const float* __restrict__ biq,
                 float* __restrict__ out,
                 int T, int F)
{
    __shared__ __align__(16) float s_in [LDSSK];   // skewed excitation span
    __shared__ __align__(16) float s_acc[LDSSK];   // skewed overlap-add acc
    __shared__ __align__(16) float s_win[WINSZ];   // Hann table

    const int blocksPerRow = F / FPB;
    const int b   = blockIdx.x / blocksPerRow;
    const int f0  = (blockIdx.x % blocksPerRow) * FPB;
    const int tid = threadIdx.x;

    // ---- Hann table + zero accumulator ------------------------------------
    const float w0 = 6.283185307179586476925f / (float)WINSZ;
    for (int t = tid; t < WINSZ; t += FPB)
        s_win[t] = 0.5f * (1.0f - cosf(w0 * (float)t));
    for (int i = tid; i < LDSSK; i += FPB)
        s_acc[i] = 0.0f;

    // ---- stage excitation span into skewed LDS ----------------------------
    // logical s_in[i] == padded-excitation column c0+i == ex column g0+i.
    const int  c0 = f0 * HOPSZ;
    const long g0 = (long)c0 - PADSZ;

    int vstart = (g0 < 0) ? (int)(-g0) : 0;
    long rem   = (long)T - g0;
    int vend   = (rem < (long)LDSIN) ? (int)rem : LDSIN;

    for (int i = tid; i < vstart; i += FPB) s_in[skew(i)] = 0.0f;
    for (int i = vend + tid; i < LDSIN; i += FPB) s_in[skew(i)] = 0.0f;

    {   // async B32 copies: 4B/lane/op, skew-compatible alignment (ASYNCcnt)
        const uint64_t gbase =
            (uint64_t)(uintptr_t)(ex + (size_t)b * (size_t)T + (size_t)(g0 + vstart));
        const uint32_t lbase0 = lds_byte_off(&s_in[0]);
        for (int i = vstart + tid; i < vend; i += FPB) {
            uint32_t loff = lbase0 + ((uint32_t)skew(i) << 2);
            uint32_t goff = (uint32_t)(i - vstart) << 2;
            asm volatile("global_load_async_to_lds_b32 %0, %1, %2"
                         :: "v"(loff), "v"(goff), "s"(gbase)
                         : "memory");
        }
        asm volatile("s_wait_asynccnt 0x0" ::: "memory");
    }
    __syncthreads();

    // ---- per-thread frame: systolic 11-stage cascade ----------------------
    const int    fr  = f0 + tid;
    const size_t gfi = (size_t)b * (size_t)F + (size_t)fr;
    const float  g   = gain[gfi];
    const size_t cb  = gfi * (size_t)(NSTG * 3);

    float na1[NSTG], na2[NSTG], y1[NSTG], y2[NSTG], xp[NSTG];
#pragma unroll
    for (int s = 0; s < NSTG; ++s) {
        na1[s] = -biq[cb + 3 * s + 1];              // a0 == 1
        na2[s] = -biq[cb + 3 * s + 2];
        y1[s] = 0.0f; y2[s] = 0.0f; xp[s] = 0.0f;
    }

    const int sbase = tid * (HOPSZ + 1);            // skew(tid*HOP)

    // iteration i: stage s consumes its input for time (i - s); all 11 stage
    // updates are independent within an iteration -> ILP ~11.
    for (int i = 0; i < WINSZ + NSTG - 1; ++i) {
        float xin = 0.0f;
        if (i < WINSZ) xin = s_in[sbase + i + (i >> 7)] * g;
        xp[0] = xin;

        float yout[NSTG];
#pragma unroll
        for (int s = 0; s < NSTG; ++s) {
            float y = __builtin_fmaf(na2[s], y2[s],
                      __builtin_fmaf(na1[s], y1[s], xp[s]));
            y2[s] = y1[s];
            y1[s] = y;
            yout[s] = y;
        }
#pragma unroll
        for (int s = NSTG - 1; s >= 1; --s) xp[s] = yout[s - 1];

        const int tt = i - (NSTG - 1);              // final-stage time index
        if (tt >= 0) {
            float v = yout[NSTG - 1] * s_win[tt];
            atomicAdd(&s_acc[sbase + tt + (tt >> 7)], v);   // ds_add_f32
        }
    }
    __syncthreads();

    // ---- coalesced flush of the block's overlap-add span ------------------
    float* __restrict__ orow = out + (size_t)b * (size_t)T;
    for (int i = tid; i < LDSIN; i += FPB) {
        const long o = (long)c0 + i - PADSZ;        // cropped output index
        if ((unsigned long)o < (unsigned long)T)
            unsafeAtomicAdd(&orow[o], s_acc[skew(i)]);  // global_atomic_add_f32
    }
}

// ---- normalization: divide by overlap-added Hann window --------------------
__global__ void __launch_bounds__(256)
lpc_norm_kernel(float* __restrict__ out, int T, int F, size_t n)
{
    const float w0 = 6.283185307179586476925f / (float)WINSZ;
    size_t stride = (size_t)gridDim.x * blockDim.x;
    for (size_t i = (size_t)blockIdx.x * blockDim.x + threadIdx.x; i < n; i += stride) {
        const int o = (int)(i % (size_t)T);
        const int p = o + PADSZ;                    // uncropped position
        float norm;
        if (p >= (WINSZ - HOPSZ) && p < F * HOPSZ) {
            norm = 2.0f;                            // 4 Hann taps 128 apart
        } else {
            norm = 0.0f;
            const int fhi = p / HOPSZ;
#pragma unroll
            for (int k = 0; k < 4; ++k) {
                const int f = fhi - k;
                if (f >= 0 && f < F) {
                    const int r = p - f * HOPSZ;    // in [0, 512)
                    norm += 0.5f * (1.0f - cosf(w0 * (float)r));
                }
            }
        }
        out[i] *= (1.0f / norm);
    }
}

extern "C" void kernel_launch(void* const* d_in, const int* in_sizes, int n_in,
                              void* d_out, int out_size, void* d_ws, size_t ws_size,
                              hipStream_t stream) {
    (void)n_in; (void)d_ws; (void)ws_size;
    const float* ex   = (const float*)d_in[0];   // (B, T)
    const float* gain = (const float*)d_in[1];   // (B, F)
    const float* biq  = (const float*)d_in[2];   // (B, F, 11, 3)
    float* out = (float*)d_out;                  // (B, T)

    const int B = 32;
    const int T = in_sizes[0] / B;               // 131072
    const int F = in_sizes[1] / B;               // 1024

    // zero the overlap-add accumulator (stream op; graph-capturable)
    hipMemsetAsync(out, 0, (size_t)out_size * sizeof(float), stream);

    const int blocks = B * (F / FPB);            // 1024 blocks x 1 wave32
    lpc_synth_kernel<<<blocks, FPB, 0, stream>>>(ex, gain, biq, out, T, F);

    const size_t n = (size_t)out_size;
    const int nblocks = (int)((n / 4 + 255) / 256);
    lpc_norm_kernel<<<nblocks, 256, 0, stream>>>(out, T, F, n);
}